// _descriptor_decoder_56160992362717
// MI455X (gfx1250) — compile-verified
//
#include <hip/hip_runtime.h>
#include <math.h>

// ---------------------------------------------------------------------------
// Fused deformable-conv + fp32-WMMA GEMM + batchnorm for MI455X (gfx1250).
//
// GEMM: sampled(M=524288, K=50) x w_def^T(K=50, N=64) via V_WMMA_F32_16X16X4_F32
//  - K padded to 52 (13 wmma k-steps of 4), exact fp32 -> matches reference.
//  - A-frag (16x4 f32): lanes 0-15 hold {K=k0,k0+1}, lanes 16-31 {k0+2,k0+3}.
//  - B-frag (4x16 f32): reg r / half-wave h holds row K=k0+r+2h, N=lane%16.
//  - C/D (16x16 f32):  reg r, half h -> M=r+8h, N=lane%16.
// ---------------------------------------------------------------------------

typedef __attribute__((ext_vector_type(2))) float v2f;
typedef __attribute__((ext_vector_type(8))) float v8f;

#define KK      25          // 5x5 deform kernel
#define KPAD    52          // GEMM K (2*25 -> pad to multiple of 4)
#define OCH     64
#define BSTRIDE 65          // LDS row stride for B matrix (bank-conflict pad)
#define SSTRIDE 56          // LDS row stride for A staging
#define HGT     256
#define WID     256
#define BATCH   8
#define HW      (HGT * WID)
#define NPIX    (BATCH * HW)    // 524288, M of the GEMM

__device__ __forceinline__ float atomAddWG(float* p, float v) {
    return __hip_atomic_fetch_add(p, v, __ATOMIC_RELAXED, __HIP_MEMORY_SCOPE_WORKGROUP);
}
__device__ __forceinline__ float atomAddDev(float* p, float v) {
    return __hip_atomic_fetch_add(p, v, __ATOMIC_RELAXED, __HIP_MEMORY_SCOPE_AGENT);
}

__global__ void zero_acc(float* g) {
    if (threadIdx.x < 128) g[threadIdx.x] = 0.f;
}

__global__ __launch_bounds__(128) void fused_deform_kernel(
    const float* __restrict__ feat,    // (B,2,H,W)
    const float* __restrict__ w_off,   // (50,2,3,3)
    const float* __restrict__ b_off,   // (50)
    const float* __restrict__ w_def,   // (64,2,5,5)
    const float* __restrict__ b_def,   // (64)
    float* __restrict__ score,         // (B,64,H,W) == d_out
    float* __restrict__ gacc)          // d_ws: [0..63]=sum, [64..127]=sumsq
{
    __shared__ float sWoff[900];               // 50*2*9
    __shared__ float sBoff[50];
    __shared__ float sB[KPAD * BSTRIDE];       // B matrix, K x N, padded rows
    __shared__ float sBdef[OCH];
    __shared__ float sStage[4][32 * SSTRIDE];  // per-wave A staging (32 pix x 52)
    __shared__ float sSum[OCH], sSqs[OCH];

    const int tid  = threadIdx.x;
    const int wave = tid >> 5;
    const int lane = tid & 31;

    // ---- cooperative weight loads into LDS -------------------------------
    for (int i = tid; i < 900; i += 128) sWoff[i] = w_off[i];
    for (int i = tid; i < 50;  i += 128) sBoff[i] = b_off[i];
    for (int i = tid; i < OCH; i += 128) sBdef[i] = b_def[i];
    for (int i = tid; i < KPAD * OCH; i += 128) {
        int kk = i >> 6, o = i & 63;
        float v = 0.f;
        if (kk < 50) { int c = kk / KK, k = kk % KK; v = w_def[(o * 2 + c) * KK + k]; }
        sB[kk * BSTRIDE + o] = v;
    }
    if (tid < OCH) { sSum[tid] = 0.f; sSqs[tid] = 0.f; }
    // zero the K-padding columns of this lane's staging row
    sStage[wave][lane * SSTRIDE + 50] = 0.f;
    sStage[wave][lane * SSTRIDE + 51] = 0.f;
    __syncthreads();

    // ---- this lane's pixel ----------------------------------------------
    const int p   = blockIdx.x * 128 + tid;   // grid exactly covers NPIX
    const int b   = p >> 16;                  // / HW (HW = 65536)
    const int rem = p & 65535;
    const int h   = rem >> 8;
    const int w   = rem & 255;
    const float* fb = feat + (size_t)b * (2 * HW);

    // 3x3 x 2ch feature patch, zero-padded
    float patch[2][9];
    #pragma unroll
    for (int ic = 0; ic < 2; ++ic)
        #pragma unroll
        for (int dy = 0; dy < 3; ++dy)
            #pragma unroll
            for (int dx = 0; dx < 3; ++dx) {
                int yy = h - 1 + dy, xx = w - 1 + dx;
                float v = 0.f;
                if (yy >= 0 && yy < HGT && xx >= 0 && xx < WID)
                    v = fb[ic * HW + yy * WID + xx];
                patch[ic][dy * 3 + dx] = v;
            }

    // ---- offset conv + bilinear sampling, staged as GEMM A rows ---------
    float* myStage = &sStage[wave][lane * SSTRIDE];
    #pragma unroll 1
    for (int k = 0; k < KK; ++k) {
        float oy = sBoff[2 * k], ox = sBoff[2 * k + 1];
        const float* wy = &sWoff[(2 * k)     * 18];
        const float* wx = &sWoff[(2 * k + 1) * 18];
        #pragma unroll
        for (int t = 0; t < 9; ++t) {
            oy = fmaf(wy[t], patch[0][t], fmaf(wy[9 + t], patch[1][t], oy));
            ox = fmaf(wx[t], patch[0][t], fmaf(wx[9 + t], patch[1][t], ox));
        }
        float py = (float)(h - 2 + k / 5) + oy;
        float px = (float)(w - 2 + k % 5) + ox;
        float y0 = floorf(py), x0 = floorf(px);
        float s0 = 0.f, s1 = 0.f;
        #pragma unroll
        for (int cy = 0; cy < 2; ++cy)
            #pragma unroll
            for (int cx = 0; cx < 2; ++cx) {
                float yy = y0 + (float)cy, xx = x0 + (float)cx;
                float wgt = (1.f - fabsf(py - yy)) * (1.f - fabsf(px - xx));
                bool valid = (yy >= 0.f) & (yy <= (float)(HGT - 1)) &
                             (xx >= 0.f) & (xx <= (float)(WID - 1));
                int yi = (int)fminf(fmaxf(yy, 0.f), (float)(HGT - 1));
                int xi = (int)fminf(fmaxf(xx, 0.f), (float)(WID - 1));
                float wv = valid ? wgt : 0.f;
                s0 = fmaf(fb[yi * WID + xi],      wv, s0);
                s1 = fmaf(fb[HW + yi * WID + xi], wv, s1);
            }
        myStage[k]      = s0;   // A row, c=0 block
        myStage[KK + k] = s1;   // A row, c=1 block
    }
    __syncthreads();

    // ---- WMMA GEMM: 2 m-tiles x 4 n-tiles, 13 k-steps -------------------
    v8f acc[2][4] = {};
    const int m    = lane & 15;
    const int half = lane >> 4;
    const float* stage0 = &sStage[wave][m * SSTRIDE];
    const float* stage1 = &sStage[wave][(16 + m) * SSTRIDE];
    #pragma unroll
    for (int kt = 0; kt < 13; ++kt) {
        const int kb = kt * 4 + 2 * half;
        v2f a0, a1;
        a0.x = stage0[kb]; a0.y = stage0[kb + 1];
        a1.x = stage1[kb]; a1.y = stage1[kb + 1];
        #pragma unroll
        for (int nt = 0; nt < 4; ++nt) {
            v2f bf;
            const int n = nt * 16 + m;
            bf.x = sB[kb * BSTRIDE + n];
            bf.y = sB[(kb + 1) * BSTRIDE + n];
            acc[0][nt] = __builtin_amdgcn_wmma_f32_16x16x4_f32(
                false, a0, false, bf, (short)0, acc[0][nt], false, false);
            acc[1][nt] = __builtin_amdgcn_wmma_f32_16x16x4_f32(
                false, a1, false, bf, (short)0, acc[1][nt], false, false);
        }
    }

    // ---- bias, store score, accumulate per-channel stats ----------------
    const int p0      = blockIdx.x * 128 + wave * 32;  // wave's base pixel
    const int bb      = p0 >> 16;                      // constant within wave chunk
    const int rembase = p0 & 65535;
    #pragma unroll
    for (int nt = 0; nt < 4; ++nt) {
        const int ch = nt * 16 + m;
        const float bias = sBdef[ch];
        float lsum = 0.f, lsqs = 0.f;
        #pragma unroll
        for (int t = 0; t < 2; ++t) {
            #pragma unroll
            for (int r = 0; r < 8; ++r) {
                float v = acc[t][nt][r] + bias;
                const int mm = r + 8 * half;
                const int rr = rembase + t * 16 + mm;  // stays inside row/batch
                score[((size_t)bb * OCH + ch) * HW + rr] = v;
                lsum += v;
                lsqs = fmaf(v, v, lsqs);
            }
        }
        atomAddWG(&sSum[ch], lsum);
        atomAddWG(&sSqs[ch], lsqs);
    }
    __syncthreads();
    if (tid < OCH) {
        atomAddDev(&gacc[tid],       sSum[tid]);
        atomAddDev(&gacc[OCH + tid], sSqs[tid]);
    }
}

__global__ void finalize_stats(const float* __restrict__ gacc,
                               const float* __restrict__ gamma,
                               const float* __restrict__ beta,
                               float* __restrict__ params)   // [0..63]=scale [64..127]=shift
{
    const int c = threadIdx.x;           // 64 threads
    const float n    = (float)NPIX;
    const float mean = gacc[c] / n;
    const float var  = gacc[OCH + c] / n - mean * mean;
    const float sc   = gamma[c] * rsqrtf(var + 1e-5f);
    params[c]        = sc;
    params[OCH + c]  = beta[c] - mean * sc;
}

__global__ __launch_bounds__(256) void normalize_kernel(
    float* __restrict__ out, const float* __restrict__ params)
{
    const size_t i = ((size_t)blockIdx.x * 256 + threadIdx.x) * 4;
    const int c = (int)((i >> 16) & 63);       // (i / HW) % 64, constant per float4
    const float sc = params[c], sh = params[OCH + c];
    float4 v = *reinterpret_cast<float4*>(out + i);
    v.x = fmaf(v.x, sc, sh);
    v.y = fmaf(v.y, sc, sh);
    v.z = fmaf(v.z, sc, sh);
    v.w = fmaf(v.w, sc, sh);
    *reinterpret_cast<float4*>(out + i) = v;
}

extern "C" void kernel_launch(void* const* d_in, const int* in_sizes, int n_in,
                              void* d_out, int out_size, void* d_ws, size_t ws_size,
                              hipStream_t stream) {
    const float* feat  = (const float*)d_in[0];
    const float* w_off = (const float*)d_in[1];
    const float* b_off = (const float*)d_in[2];
    const float* w_def = (const float*)d_in[3];
    const float* b_def = (const float*)d_in[4];
    const float* gamma = (const float*)d_in[5];
    const float* beta  = (const float*)d_in[6];
    float* out = (float*)d_out;
    float* ws  = (float*)d_ws;          // 128 acc floats + 128 param floats

    zero_acc<<<1, 128, 0, stream>>>(ws);
    fused_deform_kernel<<<NPIX / 128, 128, 0, stream>>>(
        feat, w_off, b_off, w_def, b_def, out, ws);
    finalize_stats<<<1, 64, 0, stream>>>(ws, gamma, beta, ws + 128);
    normalize_kernel<<<(NPIX * OCH) / (4 * 256), 256, 0, stream>>>(out, ws + 128);
}